// GraphColoringNet_55327768707307
// MI455X (gfx1250) — compile-verified
//
#include <hip/hip_runtime.h>

typedef __attribute__((ext_vector_type(2))) float v2f;
typedef __attribute__((ext_vector_type(8))) float v8f;

#define NN 50000
#define NE 1600000
#define DIM 64
#define ODIM 10

// Guaranteed-native fp32 scatter-add: CDNA5 GLOBAL_ATOMIC_ADD_F32, no return
// (tracked on STOREcnt; S_ENDPGM's implicit wait-idle covers completion).
__device__ __forceinline__ void atomic_add_f32(float* p, float v) {
#if defined(__AMDGCN__)
    asm volatile("global_atomic_add_f32 %0, %1, off"
                 :
                 : "v"(p), "v"(v)
                 : "memory");
#else
    (void)p; (void)v;
#endif
}

// ---- degree / normalization ----------------------------------------------
__global__ void k_init_deg(unsigned* deg, int n) {
    int i = blockIdx.x * blockDim.x + threadIdx.x;
    if (i < n) deg[i] = 1u;  // self-loop contributes 1
}

__global__ void k_count_deg(const int* __restrict__ dst, unsigned* deg, int e) {
    int i = blockIdx.x * blockDim.x + threadIdx.x;
    if (i < e) atomicAdd(&deg[dst[i]], 1u);
}

__global__ void k_dis(const unsigned* __restrict__ degc, float* dis, int n) {
    int i = blockIdx.x * blockDim.x + threadIdx.x;
    if (i < n) {
        unsigned c = degc[i];            // same buffer, same thread: safe in-place
        dis[i] = rsqrtf((float)c);
    }
}

// ---- GEMM: H[NN x 64] = X[NN x 64] @ W[64 x 64], fp32 WMMA ---------------
// One wave computes one 16x16 output tile; 4 waves/block cover N=64.
__global__ void __launch_bounds__(128) k_gemm64(const float* __restrict__ X,
                                                const float* __restrict__ W,
                                                float* __restrict__ H) {
    const int lane = threadIdx.x & 31;
    const int ntile = threadIdx.x >> 5;      // 0..3
    const int half = lane >> 4;              // 0 or 1
    const int l16  = lane & 15;
    const int mrow = blockIdx.x * 16 + l16;  // A row for this lane
    const int ncol = ntile * 16 + l16;       // B/C column for this lane

    const float* __restrict__ xrow = X + (size_t)mrow * DIM;

    v8f c = {};
#pragma unroll
    for (int k0 = 0; k0 < DIM; k0 += 4) {
        const int kb = k0 + half * 2;
        v2f a;
        a.x = xrow[kb];
        a.y = xrow[kb + 1];
        v2f b;
        b.x = W[(size_t)kb * DIM + ncol];
        b.y = W[(size_t)(kb + 1) * DIM + ncol];
        c = __builtin_amdgcn_wmma_f32_16x16x4_f32(false, a, false, b,
                                                  (short)0, c, false, false);
    }
    // D layout: VGPR v -> M = v + 8*half, N = l16
    float* __restrict__ hp = H + (size_t)(blockIdx.x * 16 + half * 8) * DIM + ncol;
#pragma unroll
    for (int v = 0; v < 8; ++v) hp[(size_t)v * DIM] = c[v];
}

// ---- AGG init: self-loop term + bias -------------------------------------
__global__ void k_init_agg(const float* __restrict__ H, const float* __restrict__ dis,
                           const float* __restrict__ bias, float* __restrict__ AGG,
                           int total) {
    int i = blockIdx.x * blockDim.x + threadIdx.x;
    if (i < total) {
        int node = i >> 6;
        int col = i & 63;
        float d = dis[node];
        AGG[i] = d * d * H[i] + bias[col];
    }
}

// ---- edge scatter: AGG[dst] += dis[src]*dis[dst] * H[src] ----------------
// 16 threads per edge, float4 per thread, native f32 atomics.
__global__ void __launch_bounds__(256) k_scatter(const int* __restrict__ src,
                                                 const int* __restrict__ dst,
                                                 const float* __restrict__ dis,
                                                 const float* __restrict__ H,
                                                 float* __restrict__ AGG) {
    long long idx = (long long)blockIdx.x * blockDim.x + threadIdx.x;
    const long long total = (long long)NE * 16;
    if (idx >= total) return;
    int e  = (int)(idx >> 4);
    int cg = (int)(idx & 15);
    int s = src[e];
    int d = dst[e];
    float nrm = dis[s] * dis[d];
    const float4 h = *(const float4*)(H + (size_t)s * DIM + cg * 4);
    float* p = AGG + (size_t)d * DIM + cg * 4;
    atomic_add_f32(p + 0, nrm * h.x);
    atomic_add_f32(p + 1, nrm * h.y);
    atomic_add_f32(p + 2, nrm * h.z);
    atomic_add_f32(p + 3, nrm * h.w);
}

// ---- ReLU ----------------------------------------------------------------
__global__ void k_relu(float* __restrict__ p, int n) {
    int i = blockIdx.x * blockDim.x + threadIdx.x;
    if (i < n) p[i] = fmaxf(p[i], 0.0f);
}

// ---- final FC: OUT[NN x 10] = H[NN x 64] @ Wfc[64 x 10] + bfc ------------
__global__ void __launch_bounds__(32) k_gemm_fc(const float* __restrict__ H,
                                                const float* __restrict__ Wfc,
                                                const float* __restrict__ bfc,
                                                float* __restrict__ OUT) {
    const int lane = threadIdx.x & 31;
    const int half = lane >> 4;
    const int l16  = lane & 15;
    const int mrow = blockIdx.x * 16 + l16;
    const bool ncolok = (l16 < ODIM);

    const float* __restrict__ xrow = H + (size_t)mrow * DIM;

    v8f c = {};
#pragma unroll
    for (int k0 = 0; k0 < DIM; k0 += 4) {
        const int kb = k0 + half * 2;
        v2f a;
        a.x = xrow[kb];
        a.y = xrow[kb + 1];
        v2f b;
        b.x = ncolok ? Wfc[(size_t)kb * ODIM + l16] : 0.0f;
        b.y = ncolok ? Wfc[(size_t)(kb + 1) * ODIM + l16] : 0.0f;
        c = __builtin_amdgcn_wmma_f32_16x16x4_f32(false, a, false, b,
                                                  (short)0, c, false, false);
    }
    if (ncolok) {
        float bb = bfc[l16];
#pragma unroll
        for (int v = 0; v < 8; ++v) {
            int row = blockIdx.x * 16 + half * 8 + v;
            OUT[(size_t)row * ODIM + l16] = c[v] + bb;
        }
    }
}

extern "C" void kernel_launch(void* const* d_in, const int* in_sizes, int n_in,
                              void* d_out, int out_size, void* d_ws, size_t ws_size,
                              hipStream_t stream) {
    (void)in_sizes; (void)n_in; (void)out_size; (void)ws_size;
    const float* x   = (const float*)d_in[0];
    const int*   ei  = (const int*)d_in[1];
    const float* W1  = (const float*)d_in[2];
    const float* b1  = (const float*)d_in[3];
    const float* W2  = (const float*)d_in[4];
    const float* b2  = (const float*)d_in[5];
    const float* Wfc = (const float*)d_in[6];
    const float* bfc = (const float*)d_in[7];
    float* out = (float*)d_out;

    const int* src = ei;        // edge_index[0]
    const int* dst = ei + NE;   // edge_index[1]

    // workspace layout
    char* ws = (char*)d_ws;
    unsigned* degc = (unsigned*)ws;                     // NN u32, reused as dis f32
    float* dis = (float*)ws;
    size_t off = ((size_t)NN * 4 + 255) & ~(size_t)255;
    float* H   = (float*)(ws + off);                    // NN*64 f32
    float* AGG = H + (size_t)NN * DIM;                  // NN*64 f32

    const int total = NN * DIM;                         // 3.2M
    const int bn = 256;
    const int gNode = (NN + bn - 1) / bn;
    const int gEdge = (NE + bn - 1) / bn;
    const int gElem = (total + bn - 1) / bn;
    const long long scTotal = (long long)NE * 16;
    const unsigned gScat = (unsigned)((scTotal + bn - 1) / bn);
    const int gTile = NN / 16;                          // 3125

    // normalization
    k_init_deg<<<gNode, bn, 0, stream>>>(degc, NN);
    k_count_deg<<<gEdge, bn, 0, stream>>>(dst, degc, NE);
    k_dis<<<gNode, bn, 0, stream>>>(degc, dis, NN);

    // layer 1
    k_gemm64<<<gTile, 128, 0, stream>>>(x, W1, H);
    k_init_agg<<<gElem, bn, 0, stream>>>(H, dis, b1, AGG, total);
    k_scatter<<<gScat, bn, 0, stream>>>(src, dst, dis, H, AGG);
    k_relu<<<gElem, bn, 0, stream>>>(AGG, total);

    // layer 2
    k_gemm64<<<gTile, 128, 0, stream>>>(AGG, W2, H);
    k_init_agg<<<gElem, bn, 0, stream>>>(H, dis, b2, AGG, total);
    k_scatter<<<gScat, bn, 0, stream>>>(src, dst, dis, H, AGG);
    k_relu<<<gElem, bn, 0, stream>>>(AGG, total);

    // final FC
    k_gemm_fc<<<gTile, 32, 0, stream>>>(AGG, Wfc, bfc, out);
}